// MultiHeadAttention_45612552683890
// MI455X (gfx1250) — compile-verified
//
#include <hip/hip_runtime.h>

typedef __attribute__((ext_vector_type(2))) float v2f;
typedef __attribute__((ext_vector_type(8))) float v8f;

#define BATCH 16
#define HEADS 16
#define IDIM  512
#define EDIM  512
#define ROWS  (BATCH * HEADS)   // 256 output rows

// ---------------------------------------------------------------------------
// Kernel 1: emb[m, e] = sum_j x[b, j, e] * W_v[h, j, e],  m = b*HEADS + h
// Softmax-column-sum identity: sum_i softmax(scores, axis=i)[i,j] == 1, so the
// whole attention block reduces to a column sum of v.
// One block per (b,h); 256 threads, 2 consecutive e per thread (float2 loads,
// fully coalesced). x and W_v are each 16 MB -> L2-resident across reuse.
// ---------------------------------------------------------------------------
__global__ __launch_bounds__(256) void emb_reduce_kernel(
    const float* __restrict__ x, const float* __restrict__ wv,
    float* __restrict__ emb)
{
    const int m  = blockIdx.x;            // 0..255
    const int b  = m / HEADS;
    const int h  = m % HEADS;
    const int e0 = threadIdx.x * 2;       // 0..510, step 2

    const float* xp = x  + (size_t)b * IDIM * EDIM + e0;
    const float* wp = wv + (size_t)h * IDIM * EDIM + e0;

    float acc0 = 0.0f, acc1 = 0.0f;
#pragma unroll 4
    for (int j = 0; j < IDIM; ++j) {
        const float2 xv = *(const float2*)(xp + (size_t)j * EDIM);
        const float2 wv2 = *(const float2*)(wp + (size_t)j * EDIM);
        acc0 = fmaf(xv.x, wv2.x, acc0);
        acc1 = fmaf(xv.y, wv2.y, acc1);
    }
    emb[(size_t)m * EDIM + e0]     = acc0;
    emb[(size_t)m * EDIM + e0 + 1] = acc1;
}

// ---------------------------------------------------------------------------
// Kernel 2: out[256,512] = emb[256,512] @ mlp_w^T[512,512] + mlp_b
// V_WMMA_F32_16X16X4_F32: D(16x16,f32) = A(16x4,f32) x B(4x16,f32) + C.
// One 16x16 output tile per wave; K=512 swept in steps of 4 (128 WMMAs/wave).
// A layout (ISA 7.12.2, 32-bit A 16x4): lanes 0-15 -> M=lane, VGPR{0,1}=K{0,1};
//                                       lanes 16-31 -> M=lane-16, VGPR{0,1}=K{2,3}.
// B mirrored (N in lanes): lane half selects K pair, lane&15 selects N.
// C/D layout: VGPR r -> M = r + 8*(lane>=16), N = lane&15.
// Grid: 64 blocks x 8 waves = 512 tiles = (256/16) x (512/16).
// ---------------------------------------------------------------------------
__global__ __launch_bounds__(256) void mlp_wmma_kernel(
    const float* __restrict__ emb, const float* __restrict__ w,
    const float* __restrict__ bias, float* __restrict__ out)
{
    const int wave  = threadIdx.x >> 5;            // 0..7
    const int lane  = threadIdx.x & 31;
    const int mtile = blockIdx.x >> 2;             // 0..15
    const int ntile = ((blockIdx.x & 3) << 3) + wave; // 0..31
    const int lhalf = lane >> 4;                   // 0 or 1
    const int l16   = lane & 15;

    const int arow = mtile * 16 + l16;             // output row (M) this lane feeds
    const int brow = ntile * 16 + l16;             // mlp_w row == output column (N)

    // lane half picks K offset {0,1} vs {2,3} within each K=4 step
    const float* ap = emb + (size_t)arow * EDIM + lhalf * 2;
    const float* bp = w   + (size_t)brow * EDIM + lhalf * 2;

    v8f c = {};
    for (int k = 0; k < EDIM; k += 4) {
        const float2 av = *(const float2*)(ap + k);
        const float2 bv = *(const float2*)(bp + k);
        v2f a;  a.x = av.x;  a.y = av.y;
        v2f bm; bm.x = bv.x; bm.y = bv.y;
        // 8 args: (neg_a, A, neg_b, B, c_mod, C, reuse_a, reuse_b)
        c = __builtin_amdgcn_wmma_f32_16x16x4_f32(
                false, a, false, bm, (short)0, c, false, false);
    }

    const float bb = bias[ntile * 16 + l16];
#pragma unroll
    for (int r = 0; r < 8; ++r) {
        const int mm = mtile * 16 + r + lhalf * 8;
        out[(size_t)mm * EDIM + ntile * 16 + l16] = c[r] + bb;
    }
}

// ---------------------------------------------------------------------------
// Launch: inputs are (x, W_q, W_k, W_v, mlp_w, mlp_b), all f32.
// W_q / W_k are mathematically dead (softmax column-sum identity).
// Workspace: emb[256 x 512] f32 = 512 KB.
// ---------------------------------------------------------------------------
extern "C" void kernel_launch(void* const* d_in, const int* in_sizes, int n_in,
                              void* d_out, int out_size, void* d_ws, size_t ws_size,
                              hipStream_t stream) {
    const float* x     = (const float*)d_in[0];
    const float* W_q   = (const float*)d_in[1];  // unused: attention collapses
    const float* W_k   = (const float*)d_in[2];  // unused: attention collapses
    const float* W_v   = (const float*)d_in[3];
    const float* mlp_w = (const float*)d_in[4];
    const float* mlp_b = (const float*)d_in[5];
    (void)W_q; (void)W_k; (void)in_sizes; (void)n_in; (void)out_size; (void)ws_size;

    float* out = (float*)d_out;
    float* emb = (float*)d_ws;   // 256*512 floats = 512 KB scratch

    emb_reduce_kernel<<<ROWS, 256, 0, stream>>>(x, W_v, emb);
    mlp_wmma_kernel<<<64, 256, 0, stream>>>(emb, mlp_w, mlp_b, out);
}